// CNN_47476568490327
// MI455X (gfx1250) — compile-verified
//
#include <hip/hip_runtime.h>

typedef __attribute__((ext_vector_type(2))) float v2f;
typedef __attribute__((ext_vector_type(8))) float v8f;

#define BN_EPS 1e-5f

// Problem sizes
#define B_    8
#define G_    96
#define C1_   8
#define C2_   16
#define P1_   48
#define P2_   24
#define NCLS_ 5

// Workspace layout (in floats). Total = 8,851,072 floats = 35.4 MB.
#define H1P_OFF 0
#define H1P_N   (B_*P1_*P1_*P1_*C1_)     // 7,077,888  pooled layer-1 activations
#define H2P_OFF (H1P_OFF + H1P_N)
#define H2P_N   (B_*P2_*P2_*P2_*C2_)     // 1,769,472  pooled layer-2 activations
#define PW2_OFF (H2P_OFF + H2P_N)
#define PW2_N   (54*64)                  // conv2 weights in WMMA B-fragment order
#define PWF_OFF (PW2_OFF + PW2_N)
#define PWF_N   (4*64)                   // fc weights in WMMA B-fragment order

// ---------------------------------------------------------------------------
// Pack W2 and Wf into V_WMMA_F32_16X16X4_F32 B-fragment order.
// B fragment for step s (K = 4s..4s+3): lane l supplies B[kbase][n], B[kbase+1][n]
// with kbase = 4s + (l>=16 ? 2 : 0), n = l & 15. Stored as [s][lane][2] floats so
// the consumer does one b64 load per lane per step.
// W2 flat layout [kd][kh][kw][cin][cout] == [k][n] with k = tap*8+cin -> identity.
// ---------------------------------------------------------------------------
__global__ void pack_weights(const float* __restrict__ W2, const float* __restrict__ Wf,
                             float* __restrict__ pw2, float* __restrict__ pwf) {
    int i = blockIdx.x * blockDim.x + threadIdx.x;
    if (i < PW2_N) {
        int s = i >> 6, w = i & 63, l = w >> 1, j = w & 1;
        int k = 4*s + ((l >= 16) ? 2 : 0) + j;
        int n = l & 15;
        pw2[i] = W2[k*16 + n];
    } else if (i < PW2_N + PWF_N) {
        int idx = i - PW2_N;
        int s = idx >> 6, w = idx & 63, l = w >> 1, j = w & 1;
        int k = 4*s + ((l >= 16) ? 2 : 0) + j;
        int n = l & 15;
        pwf[idx] = (n < NCLS_) ? Wf[k*NCLS_ + n] : 0.0f;
    }
}

// ---------------------------------------------------------------------------
// Fused conv1(3x3x3, 1->8, SAME) + BN + ReLU + 2x2x2 maxpool.
// Block = 256 threads = 4x8x8 pooled points (z,y,x). Input tile 10x18x18 in LDS.
// Cin=1 / Cout=8 gives <50% WMMA utilization and the layer is HBM-bound, so the
// VALU path is the right choice here; never materializes the pre-pool tensor.
// ---------------------------------------------------------------------------
__global__ __launch_bounds__(256) void conv1_bn_relu_pool(
    const float* __restrict__ x,  const float* __restrict__ W1,
    const float* __restrict__ b1, const float* __restrict__ g1,
    const float* __restrict__ be1,const float* __restrict__ m1,
    const float* __restrict__ v1, float* __restrict__ h1p) {
    __shared__ float xt[10*18*18];
    __shared__ float wt[216];
    __shared__ float sA[8], aA[8];

    int bx = blockIdx.x;
    int tx = bx % 6, ty = (bx/6) % 6, tz = (bx/36) % 12, b = bx/432;
    int tid = threadIdx.x;

    // Stage zero-padded input tile (Cin = 1).
    for (int i = tid; i < 10*18*18; i += 256) {
        int z = i / 324, r = i % 324, y = r / 18, xx = r % 18;
        int gz = 8*tz - 1 + z, gy = 16*ty - 1 + y, gx = 16*tx - 1 + xx;
        float v = 0.0f;
        if ((unsigned)gz < 96u && (unsigned)gy < 96u && (unsigned)gx < 96u)
            v = x[((b*96 + gz)*96 + gy)*96 + gx];
        xt[i] = v;
    }
    if (tid < 216) wt[tid] = W1[tid];                // [tap][cin=1][cout=8]
    if (tid >= 224 && tid < 232) {                   // fold BN: y = conv*s + a
        int c = tid - 224;
        float sc = g1[c] * rsqrtf(v1[c] + BN_EPS);
        sA[c] = sc;
        aA[c] = (b1[c] - m1[c]) * sc + be1[c];
    }
    __syncthreads();

    int lx = tid & 7, ly = (tid >> 3) & 7, lz = tid >> 6;
    float mx[8];
    #pragma unroll
    for (int c = 0; c < 8; ++c) mx[c] = -3.4e38f;

    #pragma unroll
    for (int q = 0; q < 8; ++q) {                    // 2x2x2 pool window
        int dz = (q >> 2) & 1, dy = (q >> 1) & 1, dx = q & 1;
        int cz = 2*lz + dz, cy = 2*ly + dy, cx = 2*lx + dx;
        float acc[8];
        #pragma unroll
        for (int c = 0; c < 8; ++c) acc[c] = 0.0f;
        for (int kd = 0; kd < 3; ++kd)
            for (int kh = 0; kh < 3; ++kh) {
                #pragma unroll
                for (int kw = 0; kw < 3; ++kw) {
                    float xv = xt[(cz + kd)*324 + (cy + kh)*18 + (cx + kw)];
                    const float* wr = &wt[((kd*3 + kh)*3 + kw)*8];
                    #pragma unroll
                    for (int c = 0; c < 8; ++c) acc[c] = fmaf(xv, wr[c], acc[c]);
                }
            }
        #pragma unroll
        for (int c = 0; c < 8; ++c) {
            float v = fmaxf(acc[c]*sA[c] + aA[c], 0.0f);
            mx[c] = fmaxf(mx[c], v);
        }
    }
    int pz = tz*4 + lz, py = ty*8 + ly, px = tx*8 + lx;
    float* o = &h1p[(((b*48 + pz)*48 + py)*48 + px)*8];
    #pragma unroll
    for (int c = 0; c < 8; ++c) o[c] = mx[c];
}

// ---------------------------------------------------------------------------
// Fused conv2(3x3x3, 8->16, SAME) + BN + ReLU + 2x2x2 maxpool as implicit GEMM
// on V_WMMA_F32_16X16X4_F32. Per wave: M = 16 conv points (two 2x2x2 pool
// cells adjacent in x), N = 16 output channels, K = 216 = 27 taps * 8 cin ->
// 54 chained WMMAs. D-layout (M 0-7 in lanes 0-15, M 8-15 in lanes 16-31)
// makes the pool a per-lane max over the 8 accumulator VGPRs, ending in one
// fully-coalesced 128B store per wave.
// ---------------------------------------------------------------------------
__global__ __launch_bounds__(128) void conv2_bn_relu_pool_wmma(
    const float* __restrict__ h1p, const float* __restrict__ pw2,
    const float* __restrict__ b2,  const float* __restrict__ g2,
    const float* __restrict__ be2, const float* __restrict__ m2,
    const float* __restrict__ v2,  float* __restrict__ h2p) {
    __shared__ float tile[4][4*4*6*8];               // per-wave [z4][y4][x6][c8]

    int lane = threadIdx.x & 31;
    int w    = threadIdx.x >> 5;
    int gw   = blockIdx.x*4 + w;
    int p0   = gw*2;                                 // first of 2 pooled points
    int px = p0 % 24, py = (p0/24) % 24, pz = (p0/576) % 24, b = p0/13824;

    // Stage zero-padded activation tile for both pool cells.
    #pragma unroll
    for (int j = 0; j < 12; ++j) {
        int u  = lane + 32*j;                        // 384 b64 units
        int r  = u / 24, qq = u % 24;
        int z  = r >> 2, y = r & 3;
        int xo = qq >> 2, c0 = (qq & 3)*2;
        int gz = 2*pz - 1 + z, gy = 2*py - 1 + y, gx = 2*px - 1 + xo;
        v2f v = {0.0f, 0.0f};
        if ((unsigned)gz < 48u && (unsigned)gy < 48u && (unsigned)gx < 48u)
            v = *(const v2f*)&h1p[((((b*48 + gz)*48 + gy)*48 + gx)*8) + c0];
        *(v2f*)&tile[w][((z*4 + y)*6 + xo)*8 + c0] = v;
    }

    // Lane -> A-matrix row (conv point): cell = m/8 picks the x-adjacent pool cell.
    int m  = lane & 15;
    int cell = m >> 3, q = m & 7;
    int dz = (q >> 2) & 1, dy = (q >> 1) & 1, dx = q & 1;
    int ch = (lane >= 16) ? 2 : 0;                   // K-half cin offset

    v8f c = {};
    const float* tw = tile[w];
    for (int kd = 0; kd < 3; ++kd)
        for (int kh = 0; kh < 3; ++kh) {
            #pragma unroll
            for (int kw = 0; kw < 3; ++kw) {
                int off = (((dz + kd)*4 + (dy + kh))*6 + (cell*2 + dx + kw))*8 + ch;
                int s   = ((kd*3 + kh)*3 + kw)*2;    // two K-steps of 4 per tap
                v2f a0 = *(const v2f*)&tw[off];          // cin ch, ch+1
                v2f b0 = *(const v2f*)&pw2[s*64 + lane*2];
                c = __builtin_amdgcn_wmma_f32_16x16x4_f32(false, a0, false, b0,
                                                          (short)0, c, false, false);
                v2f a1 = *(const v2f*)&tw[off + 4];      // cin ch+4, ch+5
                v2f b1 = *(const v2f*)&pw2[(s + 1)*64 + lane*2];
                c = __builtin_amdgcn_wmma_f32_16x16x4_f32(false, a1, false, b1,
                                                          (short)0, c, false, false);
            }
        }

    // BN + ReLU per element, then the 2x2x2 pool = max over the 8 M-rows.
    int n = lane & 15;
    float sc = g2[n] * rsqrtf(v2[n] + BN_EPS);
    float fa = (b2[n] - m2[n])*sc + be2[n];
    float pooled = -3.4e38f;
    #pragma unroll
    for (int r = 0; r < 8; ++r)
        pooled = fmaxf(pooled, fmaxf(c[r]*sc + fa, 0.0f));
    h2p[(p0 + (lane >> 4))*16 + n] = pooled;         // 128B contiguous per wave
}

// ---------------------------------------------------------------------------
// Pointwise FC 16 -> 5 as WMMA GEMM: per wave M = 16 spatial points, K = 16
// (4 WMMA steps), N padded 5 -> 16 with zeroed weights.
// ---------------------------------------------------------------------------
__global__ __launch_bounds__(128) void fc_wmma(
    const float* __restrict__ h2p, const float* __restrict__ pwf,
    const float* __restrict__ bf,  float* __restrict__ out) {
    int lane = threadIdx.x & 31;
    int gw   = blockIdx.x*4 + (threadIdx.x >> 5);
    int m0   = gw*16;
    int ch   = (lane >= 16) ? 2 : 0;
    int mr   = lane & 15;

    v8f c = {};
    #pragma unroll
    for (int s = 0; s < 4; ++s) {
        v2f a = *(const v2f*)&h2p[(m0 + mr)*16 + 4*s + ch];
        v2f b = *(const v2f*)&pwf[s*64 + lane*2];
        c = __builtin_amdgcn_wmma_f32_16x16x4_f32(false, a, false, b,
                                                  (short)0, c, false, false);
    }
    int n = lane & 15;
    if (n < NCLS_) {
        float bias = bf[n];
        #pragma unroll
        for (int r = 0; r < 8; ++r)
            out[(m0 + r + (lane >> 4)*8)*NCLS_ + n] = c[r] + bias;
    }
}

// ---------------------------------------------------------------------------
extern "C" void kernel_launch(void* const* d_in, const int* in_sizes, int n_in,
                              void* d_out, int out_size, void* d_ws, size_t ws_size,
                              hipStream_t stream) {
    (void)in_sizes; (void)n_in; (void)out_size; (void)ws_size;
    const float* x   = (const float*)d_in[0];
    const float* W1  = (const float*)d_in[1];
    const float* b1  = (const float*)d_in[2];
    const float* g1  = (const float*)d_in[3];
    const float* be1 = (const float*)d_in[4];
    const float* m1  = (const float*)d_in[5];
    const float* v1  = (const float*)d_in[6];
    const float* W2  = (const float*)d_in[7];
    const float* b2  = (const float*)d_in[8];
    const float* g2  = (const float*)d_in[9];
    const float* be2 = (const float*)d_in[10];
    const float* m2  = (const float*)d_in[11];
    const float* v2  = (const float*)d_in[12];
    const float* Wf  = (const float*)d_in[13];
    const float* bf  = (const float*)d_in[14];
    float* out = (float*)d_out;

    float* wsf = (float*)d_ws;   // requires ~35.4 MB of workspace
    float* h1p = wsf + H1P_OFF;
    float* h2p = wsf + H2P_OFF;
    float* pw2 = wsf + PW2_OFF;
    float* pwf = wsf + PWF_OFF;

    pack_weights<<<(PW2_N + PWF_N + 255)/256, 256, 0, stream>>>(W2, Wf, pw2, pwf);
    conv1_bn_relu_pool<<<3456, 256, 0, stream>>>(x, W1, b1, g1, be1, m1, v1, h1p);
    conv2_bn_relu_pool_wmma<<<13824, 128, 0, stream>>>(h1p, pw2, b2, g2, be2, m2, v2, h2p);
    fc_wmma<<<1728, 128, 0, stream>>>(h2p, pwf, bf, out);
}